// LoRARouter_91250875171188
// MI455X (gfx1250) — compile-verified
//
#include <hip/hip_runtime.h>

#define D_MODEL   4096
#define B_ROWS    8192
#define N_MODULES 7
#define N_EXPERTS 8
#define ECHUNKS   16
#define NPAD      16      // Weff padded to 16 rows so WMMA B-loads are branch-free

typedef __attribute__((ext_vector_type(2))) float v2f;
typedef __attribute__((ext_vector_type(8))) float v8f;

// ---------------------------------------------------------------------------
// Kernel 1: partial Weff[m][d] = sum_{e in chunk} Wr[m][e] * Wg[e][d]
// grid = (D_MODEL/256, ECHUNKS), block = 256.  Wg reads fully coalesced.
// ---------------------------------------------------------------------------
__global__ __launch_bounds__(256) void weff_partial(
    const float* __restrict__ Wg,   // [D_MODEL][D_MODEL]
    const float* __restrict__ Wr,   // [N_MODULES][D_MODEL]
    float* __restrict__ part)       // [ECHUNKS][N_MODULES][D_MODEL]
{
    const int d  = blockIdx.x * blockDim.x + threadIdx.x;   // 0..4095
    const int e0 = blockIdx.y * (D_MODEL / ECHUNKS);

    float acc[N_MODULES];
#pragma unroll
    for (int m = 0; m < N_MODULES; ++m) acc[m] = 0.0f;

    for (int e = e0; e < e0 + D_MODEL / ECHUNKS; ++e) {
        const float wg = Wg[(size_t)e * D_MODEL + d];
#pragma unroll
        for (int m = 0; m < N_MODULES; ++m)
            acc[m] = fmaf(Wr[m * D_MODEL + e], wg, acc[m]);
    }

    float* p = part + (size_t)blockIdx.y * (N_MODULES * D_MODEL);
#pragma unroll
    for (int m = 0; m < N_MODULES; ++m)
        p[m * D_MODEL + d] = acc[m];
}

// ---------------------------------------------------------------------------
// Kernel 2: reduce ECHUNKS partials -> WeffPad[16][4096], rows 7..15 zeroed.
// grid = NPAD*D_MODEL/256 = 256 blocks.
// ---------------------------------------------------------------------------
__global__ __launch_bounds__(256) void weff_reduce_pad(
    const float* __restrict__ part,   // [ECHUNKS][N_MODULES][D_MODEL]
    float* __restrict__ weffpad)      // [NPAD][D_MODEL]
{
    const int i = blockIdx.x * blockDim.x + threadIdx.x;    // 0..65535
    const int m = i >> 12;            // / D_MODEL
    const int d = i & (D_MODEL - 1);
    float s = 0.0f;
    if (m < N_MODULES) {
#pragma unroll
        for (int c = 0; c < ECHUNKS; ++c)
            s += part[(size_t)c * (N_MODULES * D_MODEL) + m * D_MODEL + d];
    }
    weffpad[i] = s;
}

// ---------------------------------------------------------------------------
// Kernel 3: logits via V_WMMA_F32_16X16X4_F32, fused softmax + expert masks.
// One workgroup (8 waves) per 16 rows of h.  Wave w covers K in
// [w*512, (w+1)*512); cross-wave v8f reduction through LDS.
//
// A (16x4 f32): lanes 0-15 hold {K=0,K=1}, lanes 16-31 hold {K=2,K=3},
//               M = lane & 15.
// B (4x16 f32): lanes 0-15 hold rows K=0 (v0) / K=1 (v1) at N = lane,
//               lanes 16-31 hold rows K=2 / K=3 at N = lane-16.
// D (16x16 f32): VGPR r, lane l -> [M = r + 8*(l>=16)][N = l&15].
// B rows 7..15 are zero (padded), so loads are unconditional b64.
// ---------------------------------------------------------------------------
__global__ __launch_bounds__(256) void router_wmma(
    const float* __restrict__ h,        // [B_ROWS][D_MODEL]
    const float* __restrict__ weffpad,  // [NPAD][D_MODEL]
    const float* __restrict__ noise,    // [N_MODULES][B_ROWS][N_EXPERTS]
    float* __restrict__ out)            // [N_MODULES][B_ROWS][N_EXPERTS]
{
    const int tid  = threadIdx.x;
    const int lane = tid & 31;
    const int wave = tid >> 5;
    const int row0 = blockIdx.x * 16;

    const int mn   = lane & 15;            // A-row (M) and B-col (N) for lane
    const int koff = (lane >> 4) << 1;     // 0 for lanes 0-15, 2 for 16-31

    const float* arow = h       + (size_t)(row0 + mn) * D_MODEL + koff;
    const float* brow = weffpad + (size_t)mn          * D_MODEL + koff;

    v8f acc = {0.f, 0.f, 0.f, 0.f, 0.f, 0.f, 0.f, 0.f};

    const int kbase = wave * (D_MODEL / 8);
#pragma unroll 8
    for (int k = kbase; k < kbase + D_MODEL / 8; k += 4) {
        v2f a, b;
        a.x = arow[k];
        a.y = arow[k + 1];
        b.x = brow[k];
        b.y = brow[k + 1];
        acc = __builtin_amdgcn_wmma_f32_16x16x4_f32(
            /*neg_a=*/false, a, /*neg_b=*/false, b,
            /*c_mod=*/(short)0, acc, /*reuse_a=*/false, /*reuse_b=*/false);
    }

    // ---- cross-wave reduction of the 16x16 accumulator tile ----
    __shared__ float red[8][32][8];          // 8 KB
    __shared__ float logitsS[16][8];         // 7 used per row
#pragma unroll
    for (int r = 0; r < 8; ++r) red[wave][lane][r] = acc[r];
    __syncthreads();

    if (wave == 0) {
        float s[8];
#pragma unroll
        for (int r = 0; r < 8; ++r) {
            float t = 0.0f;
#pragma unroll
            for (int w = 0; w < 8; ++w) t += red[w][lane][r];
            s[r] = t;
        }
        const int col = lane & 15;
        const int rhi = (lane >> 4) * 8;     // M offset 0 or 8
        if (col < N_MODULES) {
#pragma unroll
            for (int r = 0; r < 8; ++r) logitsS[rhi + r][col] = s[r];
        }
    }
    __syncthreads();

    // ---- per-row softmax over 7 modules + random top-k expert masks ----
    if (tid < 16) {
        const int b = row0 + tid;

        float l[N_MODULES], mx = -3.4e38f;
#pragma unroll
        for (int m = 0; m < N_MODULES; ++m) {
            l[m] = logitsS[tid][m];
            mx = fmaxf(mx, l[m]);
        }
        float ex[N_MODULES], sum = 0.0f;
#pragma unroll
        for (int m = 0; m < N_MODULES; ++m) {
            ex[m] = __expf(l[m] - mx);
            sum += ex[m];
        }
        const float inv = 1.0f / sum;

#pragma unroll
        for (int m = 0; m < N_MODULES; ++m) {
            const float p = ex[m] * inv;
            const float* nz = noise + ((size_t)m * B_ROWS + b) * N_EXPERTS;
            float nv[N_EXPERTS];
#pragma unroll
            for (int e = 0; e < N_EXPERTS; ++e) nv[e] = nz[e];

            // top-1 (stable: strict > keeps earliest index on ties,
            // matching stable descending argsort)
            int i1 = 0; float v1 = nv[0];
#pragma unroll
            for (int e = 1; e < N_EXPERTS; ++e)
                if (nv[e] > v1) { v1 = nv[e]; i1 = e; }
            // top-2
            int i2 = -1; float v2 = -3.4e38f;
#pragma unroll
            for (int e = 0; e < N_EXPERTS; ++e)
                if (e != i1 && nv[e] > v2) { v2 = nv[e]; i2 = e; }

            float* o = out + ((size_t)m * B_ROWS + b) * N_EXPERTS;
            if (p > 0.5f) {
#pragma unroll
                for (int e = 0; e < N_EXPERTS; ++e)
                    o[e] = (e == i1 || e == i2) ? 0.5f : 0.0f;
            } else {
#pragma unroll
                for (int e = 0; e < N_EXPERTS; ++e)
                    o[e] = (e == i1) ? 1.0f : 0.0f;
            }
        }
    }
}

// ---------------------------------------------------------------------------
extern "C" void kernel_launch(void* const* d_in, const int* in_sizes, int n_in,
                              void* d_out, int out_size, void* d_ws, size_t ws_size,
                              hipStream_t stream) {
    const float* h     = (const float*)d_in[0];   // [8192, 4096]
    const float* Wg    = (const float*)d_in[1];   // [4096, 4096]
    const float* Wr    = (const float*)d_in[2];   // [7, 4096]
    const float* noise = (const float*)d_in[3];   // [7, 8192, 8]
    float* out = (float*)d_out;                   // [7, 8192, 8]

    float* ws      = (float*)d_ws;
    float* weffpad = ws;                               // NPAD*4096 floats (256 KB)
    float* part    = ws + NPAD * D_MODEL;              // 16*7*4096 floats (1.8 MB)

    dim3 g1(D_MODEL / 256, ECHUNKS);
    weff_partial<<<g1, 256, 0, stream>>>(Wg, Wr, part);
    weff_reduce_pad<<<(NPAD * D_MODEL) / 256, 256, 0, stream>>>(part, weffpad);
    router_wmma<<<B_ROWS / 16, 256, 0, stream>>>(h, weffpad, noise, out);
}